// SelectiveStateSpace_24043226923321
// MI455X (gfx1250) — compile-verified
//
#include <hip/hip_runtime.h>
#include <cstdint>
#include <cstddef>

// ---------------- problem constants ----------------
#define D_STATE   16
#define D_INNER   2048
#define DT_RANK   64
#define BATCH     4
#define SEQ_LEN   4096
#define XP_COLS   96                    // DT_RANK + 2*D_STATE
#define NROWS     (BATCH * SEQ_LEN)     // 16384
#define NCHUNK    16
#define TCHUNK    (SEQ_LEN / NCHUNK)    // 256
#define NCH       (BATCH * D_INNER)     // 8192 channels
#define CARRY_STRIDE (D_STATE * NCH)    // 131072 per chunk

typedef __attribute__((ext_vector_type(2))) float v2f;
typedef __attribute__((ext_vector_type(8))) float v8f;

// =====================================================================
// Kernel 1: xp[row, 0:96] = x[row, :] @ W_x   (M=16384, N=96, K=2048)
// fp32 WMMA 16x16x4. X tile in LDS (even stride -> b64 A loads);
// W tile in LDS pair-interleaved [k/2][n][2] (stride%64==32 dwords)
// so each B fragment is ONE aligned ds_load_b64, no repacking movs.
// block = 256 thr (8 waves); each wave owns 16 rows x all 6 N-tiles.
// =====================================================================
#define KC        128
#define XT_STRIDE 130                    // even -> 8B-aligned A pairs
#define WT_STRIDE 224                    // >=192 used; 224%64==32 (bank split)

__launch_bounds__(256)
__global__ void ssm_proj_x_wmma(const float* __restrict__ x,
                                const float* __restrict__ Wx,
                                float* __restrict__ xp)
{
    __shared__ float Xt[128 * XT_STRIDE];        // 128 rows x KC (padded)
    __shared__ float Wt[(KC / 2) * WT_STRIDE];   // pair-interleaved W_x tile

    const int tid   = threadIdx.x;
    const int wave  = tid >> 5;
    const int lane  = tid & 31;
    const int lhalf = lane & 15;
    const int khalf = (lane >> 4) << 1;          // 0 or 2
    const int row_blk = blockIdx.x * 128;

    v8f c[6] = {};

    for (int k0 = 0; k0 < D_INNER; k0 += KC) {
        // stage 128x128 x-tile (coalesced along k)
        #pragma unroll
        for (int i = 0; i < (128 * KC) / 256; ++i) {
            int idx = tid + i * 256;
            int r   = idx >> 7;
            int kk  = idx & 127;
            Xt[r * XT_STRIDE + kk] = x[(size_t)(row_blk + r) * D_INNER + k0 + kk];
        }
        // stage 128x96 W_x tile, pair-interleaved: Wt[(k/2)][e][k&1]
        #pragma unroll
        for (int i = 0; i < (KC * XP_COLS) / 256; ++i) {
            int idx = tid + i * 256;
            int r   = idx / XP_COLS;              // k within chunk
            int e   = idx - r * XP_COLS;          // column
            Wt[(r >> 1) * WT_STRIDE + e * 2 + (r & 1)] =
                Wx[(size_t)(k0 + r) * XP_COLS + e];
        }
        __syncthreads();

        const int rbase = wave * 16;
        #pragma unroll 2
        for (int ks = 0; ks < KC; ks += 4) {
            const int kk = ks + khalf;            // even
            const v2f a = *(const v2f*)&Xt[(rbase + lhalf) * XT_STRIDE + kk];
            const int pbase = (kk >> 1) * WT_STRIDE + lhalf * 2;
            #pragma unroll
            for (int g = 0; g < 6; ++g) {
                const v2f b = *(const v2f*)&Wt[pbase + g * 32];
                c[g] = __builtin_amdgcn_wmma_f32_16x16x4_f32(
                    false, a, false, b, (short)0, c[g], false, false);
            }
        }
        __syncthreads();
    }

    const int row0 = row_blk + wave * 16;
    #pragma unroll
    for (int r = 0; r < 8; ++r) {
        const int m = r + ((lane >> 4) << 3);
        #pragma unroll
        for (int g = 0; g < 6; ++g) {
            xp[(size_t)(row0 + m) * XP_COLS + g * 16 + lhalf] = c[g][r];
        }
    }
}

// =====================================================================
// Kernel 2: dt[row, d] = softplus( xp[row, 0:64] @ W_dt[:, d] + b_dt[d] )
// M=16384, N=2048, K=64 (fully unrolled).
// Block = 128 rows x 512 cols; W_dt panel staged ONCE in LDS,
// pair-interleaved [k/2][j][2] (stride%64==32) -> b64 B fragments.
// All 16 A fragments cached in registers, reused across 8 col tiles.
// =====================================================================
#define WD_COLS   512
#define WD_STRIDE (WD_COLS * 2 + 32)     // 1056 dwords; 1056%64==32

__launch_bounds__(256)
__global__ void ssm_proj_dt_wmma(const float* __restrict__ xp,
                                 const float* __restrict__ Wdt,
                                 const float* __restrict__ b_dt,
                                 float* __restrict__ dt)
{
    __shared__ float Wp[(DT_RANK / 2) * WD_STRIDE];   // ~132 KB

    const int tid   = threadIdx.x;
    const int wave  = tid >> 5;
    const int lane  = tid & 31;
    const int lhalf = lane & 15;
    const int khalf = (lane >> 4) << 1;
    const int row0  = blockIdx.x * 128 + wave * 16;
    const int dbase = blockIdx.y * WD_COLS;

    // stage 64 x 512 W_dt panel, pair-interleaved
    #pragma unroll
    for (int i = 0; i < (DT_RANK * WD_COLS) / 256; ++i) {
        int idx = tid + i * 256;
        int k   = idx >> 9;                // /512
        int j   = idx & (WD_COLS - 1);
        Wp[(k >> 1) * WD_STRIDE + j * 2 + (k & 1)] =
            Wdt[(size_t)k * D_INNER + dbase + j];
    }

    // cache all A fragments (16 k-steps) in registers
    v2f afr[16];
    #pragma unroll
    for (int ks = 0; ks < 16; ++ks) {
        const int kk = ks * 4 + khalf;     // even
        afr[ks] = *(const v2f*)&xp[(size_t)(row0 + lhalf) * XP_COLS + kk];
    }
    __syncthreads();

    #pragma unroll
    for (int cg = 0; cg < 8; ++cg) {       // 8 col tiles of 64
        v8f c[4] = {};
        #pragma unroll
        for (int ks = 0; ks < 16; ++ks) {
            const int kk = ks * 4 + khalf;
            const int pbase = (kk >> 1) * WD_STRIDE + (cg * 64 + lhalf) * 2;
            #pragma unroll
            for (int g = 0; g < 4; ++g) {
                const v2f b = *(const v2f*)&Wp[pbase + g * 32];
                c[g] = __builtin_amdgcn_wmma_f32_16x16x4_f32(
                    false, afr[ks], false, b, (short)0, c[g], false, false);
            }
        }
        #pragma unroll
        for (int g = 0; g < 4; ++g) {
            const int dcol = dbase + cg * 64 + g * 16 + lhalf;
            const float bias = b_dt[dcol];
            #pragma unroll
            for (int r = 0; r < 8; ++r) {
                const int m = r + ((lane >> 4) << 3);
                float z = c[g][r] + bias;
                float sp = (z > 20.f) ? z : __logf(1.f + __expf(z));
                dt[(size_t)(row0 + m) * D_INNER + dcol] = sp;
            }
        }
    }
}

// =====================================================================
// Kernel 3 (scan pass 1): per chunk, compute carries
//   ap[n] = prod_t exp(dt*A[n]),  hloc[n] = local scan from h=0
// thread <-> channel (b,d); blockIdx.y = chunk.
// =====================================================================
__launch_bounds__(256)
__global__ void ssm_scan_pass1(const float* __restrict__ x,
                               const float* __restrict__ dtv,
                               const float* __restrict__ xp,
                               const float* __restrict__ A_log,
                               float* __restrict__ carryA,
                               float* __restrict__ carryH)
{
    const int b  = blockIdx.x >> 3;
    const int d  = ((blockIdx.x & 7) << 8) + threadIdx.x;
    const int c  = blockIdx.y;
    const int ch = b * D_INNER + d;

    float A[D_STATE], h[D_STATE], ap[D_STATE];
    #pragma unroll
    for (int n = 0; n < D_STATE; ++n) {
        A[n]  = -__expf(A_log[d * D_STATE + n]);
        h[n]  = 0.f;
        ap[n] = 1.f;
    }

    const int t0 = c * TCHUNK;
    for (int t = t0; t < t0 + TCHUNK; ++t) {
        const size_t row = (size_t)b * SEQ_LEN + t;
        if (t + 4 < t0 + TCHUNK) {
            __builtin_prefetch(&x[(row + 4) * D_INNER + d]);
            __builtin_prefetch(&dtv[(row + 4) * D_INNER + d]);
        }
        const float dtt = dtv[row * D_INNER + d];
        const float xv  = x[row * D_INNER + d];
        const float* __restrict__ xr = xp + row * XP_COLS;   // block-uniform addr
        const float dtx = dtt * xv;
        #pragma unroll
        for (int n = 0; n < D_STATE; ++n) {
            const float dA = __expf(dtt * A[n]);
            h[n]  = dA * h[n] + dtx * xr[DT_RANK + n];       // B_proj
            ap[n] *= dA;
        }
    }

    const size_t base = (size_t)c * CARRY_STRIDE;
    #pragma unroll
    for (int n = 0; n < D_STATE; ++n) {
        carryA[base + (size_t)n * NCH + ch] = ap[n];
        carryH[base + (size_t)n * NCH + ch] = h[n];
    }
}

// =====================================================================
// Kernel 4: sequential fix-up over chunks (tiny): h_init(c) per (n,ch)
// =====================================================================
__launch_bounds__(256)
__global__ void ssm_scan_fixup(const float* __restrict__ carryA,
                               const float* __restrict__ carryH,
                               float* __restrict__ hinit)
{
    const int i = blockIdx.x * blockDim.x + threadIdx.x;   // 0 .. 131071
    float h = 0.f;
    #pragma unroll
    for (int c = 0; c < NCHUNK; ++c) {
        hinit[(size_t)c * CARRY_STRIDE + i] = h;
        h = carryA[(size_t)c * CARRY_STRIDE + i] * h
          + carryH[(size_t)c * CARRY_STRIDE + i];
    }
}

// =====================================================================
// Kernel 5 (scan pass 2): re-run chunk scan from corrected h_init,
// emit y = h . C_proj + x * D_skip
// =====================================================================
__launch_bounds__(256)
__global__ void ssm_scan_pass2(const float* __restrict__ x,
                               const float* __restrict__ dtv,
                               const float* __restrict__ xp,
                               const float* __restrict__ A_log,
                               const float* __restrict__ D_skip,
                               const float* __restrict__ hinit,
                               float* __restrict__ out)
{
    const int b  = blockIdx.x >> 3;
    const int d  = ((blockIdx.x & 7) << 8) + threadIdx.x;
    const int c  = blockIdx.y;
    const int ch = b * D_INNER + d;
    const float dsk = D_skip[d];

    float A[D_STATE], h[D_STATE];
    const size_t base = (size_t)c * CARRY_STRIDE;
    #pragma unroll
    for (int n = 0; n < D_STATE; ++n) {
        A[n] = -__expf(A_log[d * D_STATE + n]);
        h[n] = hinit[base + (size_t)n * NCH + ch];
    }

    const int t0 = c * TCHUNK;
    for (int t = t0; t < t0 + TCHUNK; ++t) {
        const size_t row = (size_t)b * SEQ_LEN + t;
        if (t + 4 < t0 + TCHUNK) {
            __builtin_prefetch(&x[(row + 4) * D_INNER + d]);
            __builtin_prefetch(&dtv[(row + 4) * D_INNER + d]);
        }
        const float dtt = dtv[row * D_INNER + d];
        const float xv  = x[row * D_INNER + d];
        const float* __restrict__ xr = xp + row * XP_COLS;
        const float dtx = dtt * xv;
        float y = 0.f;
        #pragma unroll
        for (int n = 0; n < D_STATE; ++n) {
            const float dA = __expf(dtt * A[n]);
            h[n] = dA * h[n] + dtx * xr[DT_RANK + n];          // B_proj
            y += h[n] * xr[DT_RANK + D_STATE + n];             // C_proj
        }
        out[row * D_INNER + d] = y + xv * dsk;
    }
}

// =====================================================================
extern "C" void kernel_launch(void* const* d_in, const int* in_sizes, int n_in,
                              void* d_out, int out_size, void* d_ws, size_t ws_size,
                              hipStream_t stream)
{
    const float* x      = (const float*)d_in[0];
    const float* A_log  = (const float*)d_in[1];
    const float* D_skip = (const float*)d_in[2];
    const float* W_x    = (const float*)d_in[3];
    const float* W_dt   = (const float*)d_in[4];
    const float* b_dt   = (const float*)d_in[5];
    float* out = (float*)d_out;

    // workspace carve-up (floats)
    float* ws     = (float*)d_ws;
    float* xp     = ws;                                            // 16384*96
    float* dt     = xp + (size_t)NROWS * XP_COLS;                  // 16384*2048
    float* carryA = dt + (size_t)NROWS * D_INNER;                  // 16*131072
    float* carryH = carryA + (size_t)NCHUNK * CARRY_STRIDE;
    float* hinit  = carryH + (size_t)NCHUNK * CARRY_STRIDE;

    // GEMM-1: x @ W_x  (WMMA f32)
    ssm_proj_x_wmma<<<dim3(NROWS / 128), 256, 0, stream>>>(x, W_x, xp);

    // GEMM-2 + softplus: dt  (WMMA f32)
    ssm_proj_dt_wmma<<<dim3(NROWS / 128, D_INNER / WD_COLS), 256, 0, stream>>>(
        xp, W_dt, b_dt, dt);

    // chunked linear scan (diagonal transition): pass1 -> fixup -> pass2
    ssm_scan_pass1<<<dim3(NCH / 256, NCHUNK), 256, 0, stream>>>(
        x, dt, xp, A_log, carryA, carryH);
    ssm_scan_fixup<<<dim3(CARRY_STRIDE / 256), 256, 0, stream>>>(
        carryA, carryH, hinit);
    ssm_scan_pass2<<<dim3(NCH / 256, NCHUNK), 256, 0, stream>>>(
        x, dt, xp, A_log, D_skip, hinit, out);
}